// MeshPool_63909113364950
// MI455X (gfx1250) — compile-verified
//
#include <hip/hip_runtime.h>
#include <stdint.h>

// Problem constants (match reference)
#define BB 32
#define CC 512
#define EE 18000
#define TT 9000

#define TPB  128         // 4 wave32 waves per block
#define CPB  4           // channels per block == waves per block
#define TILE 1024        // staged group ids per tile buffer (4 KB)
#define NT   ((EE + TILE - 1) / TILE)   // 18 tiles (17 full + 592)

typedef float    v4f __attribute__((ext_vector_type(4)));
typedef int      v4i __attribute__((ext_vector_type(4)));
typedef unsigned v4u __attribute__((ext_vector_type(4)));

// ---------------------------------------------------------------------------
// Kernel 1: zero the per-(b,t) occupancy counters living in d_ws
// ---------------------------------------------------------------------------
__global__ void zero_counts_kernel(unsigned* __restrict__ counts) {
    int i = blockIdx.x * blockDim.x + threadIdx.x;
    if (i < BB * TT) counts[i] = 0u;
}

// ---------------------------------------------------------------------------
// Kernel 2: occupancy histogram per batch (native global_atomic_add_u32)
// ---------------------------------------------------------------------------
__global__ void count_kernel(const int* __restrict__ groups,
                             unsigned* __restrict__ counts) {
    int i = blockIdx.x * blockDim.x + threadIdx.x;
    if (i < BB * EE) {
        int b = i / EE;
        int g = groups[i];                  // in [0, TT)
        atomicAdd(&counts[b * TT + g], 1u);
    }
}

// ---------------------------------------------------------------------------
// Kernel 3: segment-mean pooling.
//   block = (b, 4-channel chunk). Wave w owns channel c0+w and a private
//   9000-float LDS accumulator row (4 * 36 KB = 144 KB; two blocks/WGP on
//   CDNA5's 320 KB LDS). Group ids are double-buffer staged with the async
//   copy path (global_load_async_to_lds_b32 + s_wait_asynccnt); fe streams
//   as non-temporal b128 loads; scatter is ds_add_f32 into the wave-private
//   row; output streams back as non-temporal b128 stores.
// ---------------------------------------------------------------------------
__global__ void __launch_bounds__(TPB)
pool_kernel(const float* __restrict__ fe,
            const int* __restrict__ groups,
            const unsigned* __restrict__ counts,
            float* __restrict__ out) {
    extern __shared__ unsigned char smem[];
    float* acc   = (float*)smem;                                    // CPB*TT f32
    int*   tiles = (int*)(smem + (size_t)CPB * TT * sizeof(float)); // 2*TILE i32
    const uint32_t dump_off = (uint32_t)(uintptr_t)(tiles + 2 * TILE);

    const int chunks = CC / CPB;                  // 128
    const int b    = blockIdx.x / chunks;
    const int c0   = (blockIdx.x % chunks) * CPB;
    const int wave = threadIdx.x >> 5;            // wave32
    const int lane = threadIdx.x & 31;

    const int*   gbase = groups + (size_t)b * EE;
    const float* ferow = fe + ((size_t)b * CC + (size_t)(c0 + wave)) * (size_t)EE;
    float*       acc_w = acc + (size_t)wave * TT;

    // stage(t, buf): exactly TILE/TPB = 8 async instructions per wave,
    // padded via a dump slot so s_wait_asynccnt immediates stay constant.
    auto stage = [&](int t, int buf) {
        const int e0  = t * TILE;
        const int len = min(TILE, EE - e0);
        int* tb = tiles + buf * TILE;
#pragma unroll
        for (int k = 0; k < TILE / TPB; ++k) {
            int  j  = (int)threadIdx.x + k * TPB;
            bool ok = j < len;
            uint32_t lds_off = ok ? (uint32_t)(uintptr_t)(tb + j) : dump_off;
            uint64_t gaddr   = (uint64_t)(uintptr_t)(gbase + e0 + (ok ? j : 0));
            asm volatile("global_load_async_to_lds_b32 %0, %1, off"
                         :: "v"(lds_off), "v"(gaddr)
                         : "memory");
        }
    };

    // kick off tile 0 copy, then zero this wave's accumulator row under it
    stage(0, 0);
    {
        const v4f z = (v4f)0.0f;
        const int nfull = TT & ~127;              // 8960
        for (int i0 = 0; i0 < nfull; i0 += 128)
            *(v4f*)(acc_w + i0 + (lane << 2)) = z;
        for (int i = nfull + lane; i < TT; i += 32) acc_w[i] = 0.f;
    }

    for (int t = 0; t < NT; ++t) {
        const int e0  = t * TILE;
        const int len = min(TILE, EE - e0);
        const bool has_next = (t + 1) < NT;

        if (has_next) {
            stage(t + 1, (t + 1) & 1);
            asm volatile("s_wait_asynccnt 8" ::: "memory");  // tile t landed
        } else {
            asm volatile("s_wait_asynccnt 0" ::: "memory");
        }
        __syncthreads();   // all waves' slices of tile t visible

        const int* tb = tiles + (t & 1) * TILE;
        if (len == TILE) {
            // fast path: compile-time trip count -> all 8 b128 loads in flight
#pragma unroll
            for (int k = 0; k < TILE / 128; ++k) {
                const int idx = k * 128 + (lane << 2);
                v4i g4 = *(const v4i*)(tb + idx);                          // ds_load_b128
                v4f v4 = __builtin_nontemporal_load((const v4f*)(ferow + e0 + idx));
                atomicAdd(&acc_w[g4.x], v4.x);                             // ds_add_f32
                atomicAdd(&acc_w[g4.y], v4.y);
                atomicAdd(&acc_w[g4.z], v4.z);
                atomicAdd(&acc_w[g4.w], v4.w);
            }
        } else {
            const int nfull = len & ~127;
            for (int i0 = 0; i0 < nfull; i0 += 128) {
                const int idx = i0 + (lane << 2);
                v4i g4 = *(const v4i*)(tb + idx);
                v4f v4 = __builtin_nontemporal_load((const v4f*)(ferow + e0 + idx));
                atomicAdd(&acc_w[g4.x], v4.x);
                atomicAdd(&acc_w[g4.y], v4.y);
                atomicAdd(&acc_w[g4.z], v4.z);
                atomicAdd(&acc_w[g4.w], v4.w);
            }
            for (int i = nfull + lane; i < len; i += 32) {
                atomicAdd(&acc_w[tb[i]], ferow[e0 + i]);
            }
        }
        __syncthreads();   // tile buffer reusable two iterations later
    }

    // write out: wave-private row, NT b128 stores, divide by max(count,1)
    const unsigned* cntrow = counts + (size_t)b * TT;
    float* orow = out + ((size_t)b * CC + (size_t)(c0 + wave)) * (size_t)TT;
    const int nfullT = TT & ~127;                 // 8960
    for (int i0 = 0; i0 < nfullT; i0 += 128) {
        const int idx = i0 + (lane << 2);
        v4f s  = *(const v4f*)(acc_w + idx);
        v4u c4 = *(const v4u*)(cntrow + idx);
        v4f r;
        r.x = s.x / fmaxf((float)c4.x, 1.0f);
        r.y = s.y / fmaxf((float)c4.y, 1.0f);
        r.z = s.z / fmaxf((float)c4.z, 1.0f);
        r.w = s.w / fmaxf((float)c4.w, 1.0f);
        __builtin_nontemporal_store(r, (v4f*)(orow + idx));
    }
    for (int i = nfullT + lane; i < TT; i += 32) {
        orow[i] = acc_w[i] / fmaxf((float)cntrow[i], 1.0f);
    }
}

// ---------------------------------------------------------------------------
extern "C" void kernel_launch(void* const* d_in, const int* in_sizes, int n_in,
                              void* d_out, int out_size, void* d_ws, size_t ws_size,
                              hipStream_t stream) {
    const float* fe     = (const float*)d_in[0];   // [B, C, E] fp32
    const int*   groups = (const int*)d_in[1];     // [B, E] int32
    float*       out    = (float*)d_out;           // [B, C, T] fp32
    unsigned*    counts = (unsigned*)d_ws;         // [B, T] u32 scratch

    {   // 1) zero occupancy counters (d_ws poisoned; re-init every call)
        int n = BB * TT;
        zero_counts_kernel<<<(n + 255) / 256, 256, 0, stream>>>(counts);
    }
    {   // 2) occupancy histogram
        int n = BB * EE;
        count_kernel<<<(n + 255) / 256, 256, 0, stream>>>(groups, counts);
    }
    {   // 3) pooled means
        dim3 grid(BB * (CC / CPB));                // 4096 workgroups
        size_t shmem = (size_t)CPB * TT * sizeof(float)    // 144000 B acc
                     + (size_t)2 * TILE * sizeof(int)      // 8192 B tiles
                     + 16;                                  // dump slot + pad
        pool_kernel<<<grid, TPB, shmem, stream>>>(fe, groups, counts, out);
    }
}